// ANIAEV_33397665694348
// MI455X (gfx1250) — compile-verified
//
#include <hip/hip_runtime.h>
#include <hip/hip_bf16.h>

// ---------------------------------------------------------------------------
// ANI AEV on MI455X (gfx1250).
// Scatter/segment-sum workload: ~140 MB of L2-resident traffic and ~78M
// scatter float-adds. Bound by L2 atomic fadd throughput, not math.
// Uses non-returning GLOBAL_ATOMIC_ADD_F32 (fire-and-forget, STOREcnt) and
// global_prefetch_b8 for the streaming edge arrays. WMMA has no contraction
// to accelerate here (per-pair rank-1 outer products of size 4x4 only).
//
// NOTE: aev_radial is defined first so the compile-loop asm snippet shows the
// scatter-atomic lowering (expect: global_atomic_add_f32, no CAS loop).
// ---------------------------------------------------------------------------

namespace {
constexpr int   kNS    = 4;
constexpr int   kNSP   = 10;
constexpr int   kRDIV  = 16;
constexpr int   kADIV  = 4;
constexpr int   kASEC  = 4;
constexpr int   kCols  = kNS * kRDIV + kNSP * kADIV * kASEC;  // 224
constexpr float kPi    = 3.14159265358979323846f;
constexpr float kRStart = 0.8f;
constexpr float kRStep  = (5.2f - 0.8f) / 16.0f;   // 0.275
constexpr float kAStart = 0.8f;
constexpr float kAStep  = (3.5f - 0.8f) / 4.0f;    // 0.675
constexpr float kZStart = kPi / 8.0f;              // pi/(2*ASEC)
constexpr float kZStep  = kPi / 4.0f;              // linspace(0,pi,5) step
constexpr float kRETA   = 16.0f;
constexpr float kAETA   = 8.0f;
constexpr int   kPrefetchDist = 16384;             // elements ahead (64KB f32)
}  // namespace

__device__ __forceinline__ void atomAddF32(float* p, float v) {
  // Lowers to non-returning global_atomic_add_f32 on gfx1250.
  unsafeAtomicAdd(p, v);
}

// ----------------------------- radial AEV ----------------------------------
// out[atom*224 + spec*16 + k] += 0.25 * exp(-16*(d - shiftR[k])^2) * sw
__global__ void aev_radial(const float* __restrict__ dist,
                           const float* __restrict__ sw,
                           const int*   __restrict__ src,
                           const int*   __restrict__ dst,
                           const int*   __restrict__ species,
                           float* __restrict__ out, int E) {
  int e = blockIdx.x * blockDim.x + threadIdx.x;
  if (e >= E) return;

  // Warm L2 ahead of the streaming arrays (global_prefetch_b8).
  if (e + kPrefetchDist < E) {
    __builtin_prefetch(dist + e + kPrefetchDist, 0, 3);
    __builtin_prefetch(sw   + e + kPrefetchDist, 0, 3);
    __builtin_prefetch(src  + e + kPrefetchDist, 0, 3);
    __builtin_prefetch(dst  + e + kPrefetchDist, 0, 3);
  }

  float d    = dist[e];
  float q    = 0.25f * sw[e];
  int   atom = src[e];
  int   spec = species[dst[e]];
  float* base = out + atom * kCols + spec * kRDIV;

#pragma unroll
  for (int k = 0; k < kRDIV; ++k) {
    float x = d - (kRStart + kRStep * (float)k);
    float t = q * __expf(-kRETA * x * x);
    atomAddF32(base + k, t);
  }
}

// ----------------------------- angular AEV ---------------------------------
// col (within species-pair block) = j*ASEC + z, offset 64 + tri*16.
__global__ void aev_angular(const float* __restrict__ angles,
                            const float* __restrict__ dist_a,
                            const float* __restrict__ sw_a,
                            const int*   __restrict__ central,
                            const int*   __restrict__ asrc,
                            const int*   __restrict__ adst,
                            const int*   __restrict__ edge_dst_a,
                            const int*   __restrict__ species,
                            float* __restrict__ out, int P) {
  int p = blockIdx.x * blockDim.x + threadIdx.x;
  if (p >= P) return;

  if (p + kPrefetchDist < P) {
    __builtin_prefetch(angles  + p + kPrefetchDist, 0, 3);
    __builtin_prefetch(central + p + kPrefetchDist, 0, 3);
    __builtin_prefetch(asrc    + p + kPrefetchDist, 0, 3);
    __builtin_prefetch(adst    + p + kPrefetchDist, 0, 3);
  }

  float ang = angles[p];
  int   ca  = central[p];
  int   e1  = asrc[p];
  int   e2  = adst[p];

  // Gathers over ~2.4 MB tables: L2-resident.
  float d12 = 0.5f * (dist_a[e1] + dist_a[e2]);
  float s   = 2.0f * sw_a[e1] * sw_a[e2];
  int   t1  = species[edge_dst_a[e1]];
  int   t2  = species[edge_dst_a[e2]];
  int   lo  = min(t1, t2);
  int   hi  = max(t1, t2);
  int   tri = (lo * (7 - lo)) / 2 + hi;           // triu index, NS=4

  float* base = out + ca * kCols + kNS * kRDIV + tri * (kADIV * kASEC);

  // factor1[z] = (0.5 + 0.5*cos(ang - shiftZ[z]))^32  (5 exact squarings)
  float f1[kASEC];
#pragma unroll
  for (int z = 0; z < kASEC; ++z) {
    float c = __cosf(ang - (kZStart + kZStep * (float)z));
    float t = 0.5f + 0.5f * c;
    t = t * t;  t = t * t;  t = t * t;  t = t * t;  t = t * t;  // ^32
    f1[z] = t;
  }

  // factor2[j] = exp(-8*(d12 - shiftA[j])^2), fold in the 2*sw1*sw2 scale.
  float f2[kADIV];
#pragma unroll
  for (int j = 0; j < kADIV; ++j) {
    float x = d12 - (kAStart + kAStep * (float)j);
    f2[j] = s * __expf(-kAETA * x * x);
  }

#pragma unroll
  for (int j = 0; j < kADIV; ++j) {
#pragma unroll
    for (int z = 0; z < kASEC; ++z) {
      atomAddF32(base + j * kASEC + z, f2[j] * f1[z]);
    }
  }
}

// --------------------------- zero the output -------------------------------
__global__ void aev_zero(float* __restrict__ out, int n) {
  int i = blockIdx.x * blockDim.x + threadIdx.x;
  if (i < n) out[i] = 0.0f;
}

// ------------------------------ launcher -----------------------------------
extern "C" void kernel_launch(void* const* d_in, const int* in_sizes, int n_in,
                              void* d_out, int out_size, void* d_ws, size_t ws_size,
                              hipStream_t stream) {
  const int*   species    = (const int*)  d_in[0];
  const float* dist_r     = (const float*)d_in[1];
  const float* sw_r       = (const float*)d_in[2];
  const int*   edge_src   = (const int*)  d_in[3];
  const int*   edge_dst_r = (const int*)  d_in[4];
  const float* angles     = (const float*)d_in[5];
  const float* dist_a     = (const float*)d_in[6];
  const int*   central    = (const int*)  d_in[7];
  const int*   angle_src  = (const int*)  d_in[8];
  const int*   angle_dst  = (const int*)  d_in[9];
  const float* sw_a       = (const float*)d_in[10];
  const int*   edge_dst_a = (const int*)  d_in[11];
  float* out = (float*)d_out;

  const int E = in_sizes[1];   // 1,600,000 radial edges
  const int P = in_sizes[5];   // 3,300,000 angle pairs

  constexpr int kBlk = 256;    // 8 wave32 per block

  aev_zero<<<(out_size + kBlk - 1) / kBlk, kBlk, 0, stream>>>(out, out_size);

  aev_radial<<<(E + kBlk - 1) / kBlk, kBlk, 0, stream>>>(
      dist_r, sw_r, edge_src, edge_dst_r, species, out, E);

  aev_angular<<<(P + kBlk - 1) / kBlk, kBlk, 0, stream>>>(
      angles, dist_a, sw_a, central, angle_src, angle_dst, edge_dst_a,
      species, out, P);
}